// FeaStNet_17111149707375
// MI455X (gfx1250) — compile-verified
//
#include <hip/hip_runtime.h>
#include <math.h>

typedef __attribute__((ext_vector_type(2))) float v2f;
typedef __attribute__((ext_vector_type(8))) float v8f;

#define HEADS 8
#define EPSV 1e-5f

// =====================================================================
// WMMA f32 GEMM:  C[M,Nc] = A[M,K] @ B[K,Nc]  (+optional bias, relu)
// Row-major fp32. M % 64 == 0 (caller pads), K % 4 == 0, Nc % 64 == 0.
// One wave per 64x64 macro-tile: 4x4 grid of 16x16 WMMA accumulators,
// 16 x V_WMMA_F32_16X16X4_F32 per K-step, 1 float/lane of fragment
// traffic per WMMA (vs 4 for a single-tile wave) -> 4x less L2 traffic.
// Fragment layouts per CDNA5 ISA 7.12.2:
//   A 16x4 : lanes 0-15 / 16-31 both carry M=0..15; VGPR0={K0,K2}, VGPR1={K1,K3}
//   B 4x16 : mirror (lanes carry N=0..15; half-wave selects K pair)
//   C/D    : VGPR r -> M = r (lanes 0-15), M = r+8 (lanes 16-31); N = lane&15
// =====================================================================
__global__ __launch_bounds__(256) void gemm_f32_wmma(
    const float* __restrict__ A, const float* __restrict__ B,
    float* __restrict__ C, const float* __restrict__ bias,
    int M, int K, int Nc, int relu)
{
    const int wid  = (blockIdx.x * blockDim.x + threadIdx.x) >> 5;
    const int lane = threadIdx.x & 31;
    const int tilesN = Nc >> 6;             // 64-wide macro tiles
    const int tilesM = M >> 6;
    if (wid >= tilesM * tilesN) return;     // wave-uniform: EXEC all-1 inside
    const int tm  = wid / tilesN;
    const int tn  = wid - tm * tilesN;
    const int l15 = lane & 15;
    const int hi  = lane >> 4;              // half-wave selects K pair
    const int rowBase = (tm << 6) + l15;    // + 16*i for sub-tile i
    const int colBase = (tn << 6) + l15;    // + 16*j for sub-tile j

    v8f acc[4][4];
#pragma unroll
    for (int i = 0; i < 4; ++i)
#pragma unroll
        for (int j = 0; j < 4; ++j)
            acc[i][j] = (v8f){0.f, 0.f, 0.f, 0.f, 0.f, 0.f, 0.f, 0.f};

    for (int k0 = 0; k0 < K; k0 += 4) {
        const int ka = k0 + (hi << 1);
        v2f a[4], b[4];
#pragma unroll
        for (int i = 0; i < 4; ++i) {
            const float* __restrict__ Ar = A + (size_t)(rowBase + (i << 4)) * K + ka;
            a[i].x = Ar[0];
            a[i].y = Ar[1];
        }
#pragma unroll
        for (int j = 0; j < 4; ++j) {
            const float* __restrict__ Bc = B + (size_t)ka * Nc + colBase + (j << 4);
            b[j].x = Bc[0];
            b[j].y = Bc[Nc];
        }
#pragma unroll
        for (int i = 0; i < 4; ++i)
#pragma unroll
            for (int j = 0; j < 4; ++j)
                acc[i][j] = __builtin_amdgcn_wmma_f32_16x16x4_f32(
                    false, a[i], false, b[j], (short)0, acc[i][j], false, false);
    }

#pragma unroll
    for (int j = 0; j < 4; ++j) {
        const int col = colBase + (j << 4);
        const float bb = bias ? bias[col] : 0.f;
#pragma unroll
        for (int i = 0; i < 4; ++i) {
#pragma unroll
            for (int r = 0; r < 8; ++r) {
                const int row = (tm << 6) + (i << 4) + r + (hi << 3);
                float v = acc[i][j][r] + bb;
                if (relu) v = fmaxf(v, 0.f);
                C[(size_t)row * Nc + col] = v;
            }
        }
    }
}

// ---------------------------------------------------------------------
// XU = X @ U  (N x K  @  K x 8).  One wave per node: lanes stride K
// (coalesced row streaming), 8 head accumulators reduced via shuffles.
// ---------------------------------------------------------------------
__global__ __launch_bounds__(256) void xu_kernel(
    const float* __restrict__ X, const float* __restrict__ U,
    float* __restrict__ XU, int N, int K)
{
    const int wid  = (blockIdx.x * blockDim.x + threadIdx.x) >> 5;
    const int lane = threadIdx.x & 31;
    if (wid >= N) return;
    float acc[HEADS];
#pragma unroll
    for (int h = 0; h < HEADS; ++h) acc[h] = 0.f;
    const float* __restrict__ xr = X + (size_t)wid * K;
    for (int k = lane; k < K; k += 32) {
        const float xv = xr[k];
        const float* __restrict__ ur = U + (size_t)k * HEADS;
#pragma unroll
        for (int h = 0; h < HEADS; ++h) acc[h] += xv * ur[h];
    }
#pragma unroll
    for (int h = 0; h < HEADS; ++h)
#pragma unroll
        for (int o = 16; o > 0; o >>= 1)
            acc[h] += __shfl_down(acc[h], o, 32);
    if (lane == 0) {
#pragma unroll
        for (int h = 0; h < HEADS; ++h) XU[(size_t)wid * HEADS + h] = acc[h];
    }
}

// ---------------------------------------------------------------------
// Per-edge FeaSt message + scatter. One wave32 per edge (incl. self loops).
// q = softmax_h(XU[s]-XU[d]+c); agg[d] += sum_h q_h * XW[s, h, :]
// ---------------------------------------------------------------------
__global__ __launch_bounds__(256) void feast_edge_kernel(
    const int* __restrict__ ei, int E, int N,
    const float* __restrict__ XW, const float* __restrict__ XU,
    const float* __restrict__ Chead, float* __restrict__ agg, int Cout)
{
    const int wid  = (blockIdx.x * blockDim.x + threadIdx.x) >> 5;
    const int lane = threadIdx.x & 31;
    const int nE = E + N;
    if (wid >= nE) return;                  // wave-uniform exit

    int s, d;
    if (wid < E) { s = ei[wid]; d = ei[E + wid]; }
    else         { s = wid - E; d = s; }    // self loop

    // prefetch the gathered weighted-feature row while softmax is computed
    const float* __restrict__ row = XW + (size_t)s * HEADS * Cout;
    __builtin_prefetch(row + lane * 16, 0, 3);

    // 8-head softmax, computed redundantly by all lanes (broadcast loads)
    float q[HEADS];
    float mx = -1e30f;
#pragma unroll
    for (int h = 0; h < HEADS; ++h) {
        const float t = XU[(size_t)s * HEADS + h] - XU[(size_t)d * HEADS + h] + Chead[h];
        q[h] = t;
        mx = fmaxf(mx, t);
    }
    float sum = 0.f;
#pragma unroll
    for (int h = 0; h < HEADS; ++h) { q[h] = __expf(q[h] - mx); sum += q[h]; }
    const float inv = 1.f / sum;
#pragma unroll
    for (int h = 0; h < HEADS; ++h) q[h] *= inv;

    float* __restrict__ out = agg + (size_t)d * Cout;
    for (int c = lane; c < Cout; c += 32) {
        float m = 0.f;
#pragma unroll
        for (int h = 0; h < HEADS; ++h) m += q[h] * row[(size_t)h * Cout + c];
        atomicAdd(&out[c], m);
    }
}

// ---------------------------------------------------------------------
// Per-channel sum & sumsq over rows (one block per channel, LDS tree)
// stats[c] = sum, stats[C+c] = sumsq
// ---------------------------------------------------------------------
__global__ __launch_bounds__(256) void col_stats_kernel(
    const float* __restrict__ Y, int N, int C, float* __restrict__ stats)
{
    const int c = blockIdx.x;
    __shared__ float s1[256];
    __shared__ float s2[256];
    float a = 0.f, b = 0.f;
    for (int r = threadIdx.x; r < N; r += blockDim.x) {
        const float v = Y[(size_t)r * C + c];
        a += v; b += v * v;
    }
    s1[threadIdx.x] = a; s2[threadIdx.x] = b;
    __syncthreads();
    for (int s = blockDim.x >> 1; s > 0; s >>= 1) {
        if (threadIdx.x < s) {
            s1[threadIdx.x] += s1[threadIdx.x + s];
            s2[threadIdx.x] += s2[threadIdx.x + s];
        }
        __syncthreads();
    }
    if (threadIdx.x == 0) { stats[c] = s1[0]; stats[C + c] = s2[0]; }
}

// ---------------------------------------------------------------------
// agg[i,c] = agg[i,c]/cnt[i] + bias[c]     (in place)
// ---------------------------------------------------------------------
__global__ __launch_bounds__(256) void div_bias_kernel(
    float* __restrict__ Agg, const float* __restrict__ cnt,
    const float* __restrict__ bias, int N, int C)
{
    const size_t idx = (size_t)blockIdx.x * blockDim.x + threadIdx.x;
    if (idx >= (size_t)N * C) return;
    const int i = (int)(idx / C);
    const int c = (int)(idx - (size_t)i * C);
    Agg[idx] = Agg[idx] / cnt[i] + bias[c];
}

// ---------------------------------------------------------------------
// Out = relu?((Y - mean) * rsqrt(var+eps) * g + b)
// ---------------------------------------------------------------------
__global__ __launch_bounds__(256) void bn_relu_kernel(
    const float* __restrict__ Y, const float* __restrict__ stats,
    const float* __restrict__ g, const float* __restrict__ b,
    float* __restrict__ Out, int N, int C, int relu)
{
    const size_t idx = (size_t)blockIdx.x * blockDim.x + threadIdx.x;
    if (idx >= (size_t)N * C) return;
    const int c = (int)(idx % C);
    const float invN = 1.f / (float)N;
    const float mean = stats[c] * invN;
    const float var  = stats[C + c] * invN - mean * mean;
    float v = (Y[idx] - mean) * rsqrtf(var + EPSV) * g[c] + b[c];
    if (relu) v = fmaxf(v, 0.f);
    Out[idx] = v;
}

// ---------------------------------------------------------------------
// Entry: h = relu(concat(bn(pos), x) @ lin0_W + lin0_b)   (4 -> 16)
// ---------------------------------------------------------------------
__global__ __launch_bounds__(256) void pre_kernel(
    const float* __restrict__ pos, const float* __restrict__ x,
    const float* __restrict__ stats, const float* __restrict__ g0,
    const float* __restrict__ b0, const float* __restrict__ W,
    const float* __restrict__ bias, float* __restrict__ H, int N)
{
    const int i = blockIdx.x * blockDim.x + threadIdx.x;
    if (i >= N) return;
    float in4[4];
    const float invN = 1.f / (float)N;
#pragma unroll
    for (int c = 0; c < 2; ++c) {
        const float m = stats[c] * invN;
        const float v = stats[2 + c] * invN - m * m;
        in4[c] = (pos[(size_t)i * 2 + c] - m) * rsqrtf(v + EPSV) * g0[c] + b0[c];
    }
    in4[2] = x[(size_t)i * 2 + 0];
    in4[3] = x[(size_t)i * 2 + 1];
#pragma unroll
    for (int o = 0; o < 16; ++o) {
        float a = bias[o];
#pragma unroll
        for (int k = 0; k < 4; ++k) a += in4[k] * W[k * 16 + o];
        H[(size_t)i * 16 + o] = fmaxf(a, 0.f);
    }
}

// ---------------------------------------------------------------------
// Exit: out = H(64) @ lin2_W(64x2) + lin2_b
// ---------------------------------------------------------------------
__global__ __launch_bounds__(256) void lin2_kernel(
    const float* __restrict__ H, const float* __restrict__ W,
    const float* __restrict__ b, float* __restrict__ out, int N)
{
    const int i = blockIdx.x * blockDim.x + threadIdx.x;
    if (i >= N) return;
    const float* __restrict__ hr = H + (size_t)i * 64;
    float a0 = b[0], a1 = b[1];
    for (int k = 0; k < 64; ++k) {
        const float hv = hr[k];
        a0 += hv * W[k * 2 + 0];
        a1 += hv * W[k * 2 + 1];
    }
    out[(size_t)i * 2 + 0] = a0;
    out[(size_t)i * 2 + 1] = a1;
}

// ---------------------------------------------------------------------
// utilities
// ---------------------------------------------------------------------
__global__ __launch_bounds__(256) void fill_kernel(float* __restrict__ p, float v, size_t n)
{
    for (size_t idx = (size_t)blockIdx.x * blockDim.x + threadIdx.x; idx < n;
         idx += (size_t)gridDim.x * blockDim.x)
        p[idx] = v;
}

__global__ __launch_bounds__(256) void count_kernel(
    const int* __restrict__ ei, float* __restrict__ cnt, int E)
{
    const int e = blockIdx.x * blockDim.x + threadIdx.x;
    if (e >= E) return;
    atomicAdd(&cnt[ei[E + e]], 1.0f);
}

// =====================================================================
// host launcher
// =====================================================================
static inline int cdiv(long long a, long long b) { return (int)((a + b - 1) / b); }

extern "C" void kernel_launch(void* const* d_in, const int* in_sizes, int n_in,
                              void* d_out, int out_size, void* d_ws, size_t ws_size,
                              hipStream_t stream)
{
    const int N = in_sizes[0] / 2;
    const int E = in_sizes[2] / 2;
    const int Mpad = (N + 63) & ~63;        // GEMM M padded to 64-row macro tiles

    const float* pos    = (const float*)d_in[0];
    const float* x      = (const float*)d_in[1];
    const int*   ei     = (const int*)  d_in[2];
    const float* g0     = (const float*)d_in[3];
    const float* b0     = (const float*)d_in[4];
    const float* lin0_W = (const float*)d_in[5];
    const float* lin0_b = (const float*)d_in[6];
    const float* lin1_W = (const float*)d_in[43];
    const float* lin1_b = (const float*)d_in[44];
    const float* lin2_W = (const float*)d_in[45];
    const float* lin2_b = (const float*)d_in[46];

    const int ci[6] = {16, 32, 64, 128, 256, 128};
    const int co[6] = {32, 64, 128, 256, 128, 128};

    // workspace carve (floats)
    float* ws = (float*)d_ws;
    size_t off = 0;
    float* hA    = ws + off; off += (size_t)Mpad * 256;
    float* hB    = ws + off; off += (size_t)Mpad * 256;
    float* XW    = ws + off; off += (size_t)Mpad * 2048;
    float* XU    = ws + off; off += (size_t)N * HEADS;
    float* agg   = ws + off; off += (size_t)N * 256;
    float* cnt   = ws + off; off += (size_t)N;
    float* stats = ws + off; off += 512;

    const int TPB = 256;

    // degree counts (layer-invariant): cnt = 1 (self loop) + in-degree
    fill_kernel<<<cdiv(N, TPB), TPB, 0, stream>>>(cnt, 1.0f, (size_t)N);
    count_kernel<<<cdiv(E, TPB), TPB, 0, stream>>>(ei, cnt, E);

    // entry: BN(pos) stats, then fused concat+lin0+relu -> hA (N x 16)
    col_stats_kernel<<<2, TPB, 0, stream>>>(pos, N, 2, stats);
    pre_kernel<<<cdiv(N, TPB), TPB, 0, stream>>>(pos, x, stats, g0, b0,
                                                 lin0_W, lin0_b, hA, N);
    if (Mpad > N)   // zero pad rows so WMMA macro tiles read zeros
        fill_kernel<<<cdiv((long long)(Mpad - N) * 16, TPB), TPB, 0, stream>>>(
            hA + (size_t)N * 16, 0.f, (size_t)(Mpad - N) * 16);

    float* hin = hA;
    float* hout = hB;
    for (int l = 0; l < 6; ++l) {
        const float* W  = (const float*)d_in[7 + 6 * l];
        const float* U  = (const float*)d_in[8 + 6 * l];
        const float* Ch = (const float*)d_in[9 + 6 * l];
        const float* Bi = (const float*)d_in[10 + 6 * l];
        const float* g  = (const float*)d_in[11 + 6 * l];
        const float* b  = (const float*)d_in[12 + 6 * l];
        const int K    = ci[l];
        const int Cout = co[l];
        const int Nc   = HEADS * Cout;

        // XW = h @ W via f32 WMMA (64x64 macro tiles)
        {
            const long long waves = (long long)(Mpad / 64) * (Nc / 64);
            gemm_f32_wmma<<<cdiv(waves * 32, TPB), TPB, 0, stream>>>(
                hin, W, XW, nullptr, Mpad, K, Nc, 0);
        }
        // XU = h @ U
        xu_kernel<<<cdiv((long long)N * 32, TPB), TPB, 0, stream>>>(hin, U, XU, N, K);

        // scatter
        fill_kernel<<<cdiv((long long)N * Cout, TPB), TPB, 0, stream>>>(
            agg, 0.0f, (size_t)N * Cout);
        {
            const long long waves = (long long)E + N;
            feast_edge_kernel<<<cdiv(waves * 32, TPB), TPB, 0, stream>>>(
                ei, E, N, XW, XU, Ch, agg, Cout);
        }

        // mean over neighbors + conv bias, then BN + ReLU
        div_bias_kernel<<<cdiv((long long)N * Cout, TPB), TPB, 0, stream>>>(
            agg, cnt, Bi, N, Cout);
        col_stats_kernel<<<Cout, TPB, 0, stream>>>(agg, N, Cout, stats);
        bn_relu_kernel<<<cdiv((long long)N * Cout, TPB), TPB, 0, stream>>>(
            agg, stats, g, b, hout, N, Cout, 1);
        if (Mpad > N)   // keep pad rows zero for the next layer's GEMM
            fill_kernel<<<cdiv((long long)(Mpad - N) * Cout, TPB), TPB, 0, stream>>>(
                hout + (size_t)N * Cout, 0.f, (size_t)(Mpad - N) * Cout);

        float* t = hin; hin = hout; hout = t;
    }

    // lin1: 128 -> 64, bias + relu, via WMMA
    {
        const long long waves = (long long)(Mpad / 64) * (64 / 64);
        gemm_f32_wmma<<<cdiv(waves * 32, TPB), TPB, 0, stream>>>(
            hin, lin1_W, hout, lin1_b, Mpad, 128, 64, 1);
    }
    // lin2: 64 -> 2
    lin2_kernel<<<cdiv(N, TPB), TPB, 0, stream>>>(hout, lin2_W, lin2_b,
                                                  (float*)d_out, N);
    (void)n_in; (void)out_size; (void)ws_size; (void)off;
}